// SAGEConv_13005160973068
// MI455X (gfx1250) — compile-verified
//
#include <hip/hip_runtime.h>

typedef __attribute__((ext_vector_type(2))) float v2f;
typedef __attribute__((ext_vector_type(4))) float v4f;
typedef __attribute__((ext_vector_type(8))) float v8f;

#define NN 100000
#define DI 128
#define DO 128
#define LDA 260  // LDS row stride in floats: 256 data + 4 pad (4 mod 64 banks -> conflict-free)

// ---------------------------------------------------------------------------
// Kernel 1: zero the scatter accumulator (aliased onto d_out) and deg (in ws)
// ---------------------------------------------------------------------------
__global__ __launch_bounds__(256) void sage_init(float* __restrict__ out,
                                                 float* __restrict__ deg) {
  int tid = blockIdx.x * blockDim.x + threadIdx.x;
  int stride = gridDim.x * blockDim.x;
  v4f z = {0.f, 0.f, 0.f, 0.f};
  v4f* o4 = (v4f*)out;
  const int total4 = NN * DO / 4;
  for (int i = tid; i < total4; i += stride) o4[i] = z;
  for (int i = tid; i < NN; i += stride) deg[i] = 0.f;
}

// ---------------------------------------------------------------------------
// Kernel 2: one wave per edge. 32 lanes x float4 gather of h[src], then
// 4 global_atomic_add_f32 per lane into msg[dst] (L2-resident accumulator).
// ---------------------------------------------------------------------------
__global__ __launch_bounds__(256) void sage_scatter(
    const float* __restrict__ h, const long long* __restrict__ src,
    const long long* __restrict__ dst, float* __restrict__ msg,
    float* __restrict__ deg, int n_edges) {
  int e = (int)((blockIdx.x * 256u + threadIdx.x) >> 5);
  int lane = threadIdx.x & 31;
  if (e >= n_edges) return;
  int s = (int)src[e];
  int d = (int)dst[e];
  v4f v = *(const v4f*)(h + (long)s * DI + lane * 4);
  float* p = msg + (long)d * DI + lane * 4;
  __hip_atomic_fetch_add(p + 0, v.x, __ATOMIC_RELAXED, __HIP_MEMORY_SCOPE_AGENT);
  __hip_atomic_fetch_add(p + 1, v.y, __ATOMIC_RELAXED, __HIP_MEMORY_SCOPE_AGENT);
  __hip_atomic_fetch_add(p + 2, v.z, __ATOMIC_RELAXED, __HIP_MEMORY_SCOPE_AGENT);
  __hip_atomic_fetch_add(p + 3, v.w, __ATOMIC_RELAXED, __HIP_MEMORY_SCOPE_AGENT);
  if (lane == 0)
    __hip_atomic_fetch_add(deg + d, 1.0f, __ATOMIC_RELAXED, __HIP_MEMORY_SCOPE_AGENT);
}

// ---------------------------------------------------------------------------
// Kernel 3: fused normalize + GEMM with V_WMMA_F32_16X16X4_F32.
// Block = 256 threads (8 waves) -> 16 rows x 128 cols of output.
// `out` holds msg_sum on entry; rows are staged (scaled by 1/deg) into LDS
// alongside h rows (A tile = 16 x 256), then overwritten with the result.
// ---------------------------------------------------------------------------
__global__ __launch_bounds__(256) void sage_gemm(
    const float* __restrict__ h, const float* __restrict__ W,
    const float* __restrict__ bias, const float* __restrict__ deg,
    float* __restrict__ out) {
  __shared__ float A[16 * LDA];
  const int m0 = blockIdx.x * 16;
  const int tid = threadIdx.x;

  // stage h rows -> A[:, 0:128]
  for (int idx = tid; idx < 512; idx += 256) {
    int row = idx >> 5;
    int c4 = (idx & 31) << 2;
    v4f v = *(const v4f*)(h + (long)(m0 + row) * DI + c4);
    *(v4f*)&A[row * LDA + c4] = v;
  }
  // stage normalized msg rows -> A[:, 128:256]  (reads out == msg_sum)
  for (int idx = tid; idx < 512; idx += 256) {
    int row = idx >> 5;
    int c4 = (idx & 31) << 2;
    float dg = deg[m0 + row];
    float sc = dg > 0.f ? 1.f / dg : 0.f;
    v4f v = *(const v4f*)(out + (long)(m0 + row) * DO + c4);
    v.x *= sc; v.y *= sc; v.z *= sc; v.w *= sc;
    *(v4f*)&A[row * LDA + DI + c4] = v;
  }
  __syncthreads();

  const int wave = tid >> 5;
  const int lane = tid & 31;
  const int n0 = wave << 4;       // this wave's 16-column slab
  const int nn = lane & 15;       // N within tile
  const int khalf = (lane < 16) ? 0 : 2;  // f32 16x16x4 A/B lane->K mapping

  float bv = bias[n0 + nn];
  v8f acc;
#pragma unroll
  for (int r = 0; r < 8; r++) acc[r] = bv;

  const float* Ap = &A[(lane & 15) * LDA + khalf];  // A[m][k+khalf], m=lane%16
  const float* Wp = W + khalf * DO + n0 + nn;       // W[k+khalf][n0+nn]

#pragma unroll 8
  for (int k = 0; k < 256; k += 4) {
    v2f a = *(const v2f*)(Ap + k);  // ds_load_b64: A[m][k+khalf], A[m][k+khalf+1]
    v2f bb;
    bb.x = Wp[k * DO];              // W[k+khalf  ][n0+nn]
    bb.y = Wp[k * DO + DO];         // W[k+khalf+1][n0+nn]
    acc = __builtin_amdgcn_wmma_f32_16x16x4_f32(
        false, a, false, bb, (short)0, acc, false, false);
  }

  // D layout: VGPR r, lane l -> M = r + (l<16 ? 0 : 8), N = l&15
  const int mb = (lane < 16) ? 0 : 8;
  float* op = out + (long)(m0 + mb) * DO + n0 + nn;
#pragma unroll
  for (int r = 0; r < 8; r++) op[r * DO] = acc[r];
}

// ---------------------------------------------------------------------------
extern "C" void kernel_launch(void* const* d_in, const int* in_sizes, int n_in,
                              void* d_out, int out_size, void* d_ws, size_t ws_size,
                              hipStream_t stream) {
  const float* h = (const float*)d_in[0];
  const long long* src = (const long long*)d_in[1];
  const long long* dst = (const long long*)d_in[2];
  const float* W = (const float*)d_in[3];
  const float* b = (const float*)d_in[4];
  float* out = (float*)d_out;    // doubles as msg_sum accumulator
  float* deg = (float*)d_ws;     // 100000 floats = 400 KB scratch
  const int n_edges = in_sizes[1];

  sage_init<<<2048, 256, 0, stream>>>(out, deg);
  int sgrid = (n_edges + 7) / 8;  // 8 edges (waves) per 256-thread block
  sage_scatter<<<sgrid, 256, 0, stream>>>(h, src, dst, out, deg, n_edges);
  sage_gemm<<<NN / 16, 256, 0, stream>>>(h, W, b, deg, out);
}